// AdaptiveAttention_12378095748015
// MI455X (gfx1250) — compile-verified
//
#include <hip/hip_runtime.h>
#include <hip/hip_bf16.h>
#include <math.h>

// Problem constants (match reference)
#define BB 32
#define CC 256
#define CR 16
#define HH 64
#define WW 64
#define HWN (HH * WW)          // 4096
#define GN_EPS 1e-5f

typedef __attribute__((ext_vector_type(2))) float v2f;
typedef __attribute__((ext_vector_type(8))) float v8f;

__device__ __forceinline__ float sigmoidf_(float v) { return 1.0f / (1.0f + expf(-v)); }

// ---------------------------------------------------------------------------
// Kernel 1: per-(b,c) plane mean. One block per 4096-float contiguous plane.
// Fully coalesced float4 reads; deterministic LDS tree reduction.
// ---------------------------------------------------------------------------
__global__ __launch_bounds__(256) void k_plane_mean(const float* __restrict__ x,
                                                    float* __restrict__ p /*[B*C]*/) {
    const float4* xp = (const float4*)(x + (size_t)blockIdx.x * HWN);
    float s = 0.0f;
    #pragma unroll
    for (int i = 0; i < 4; ++i) {              // 1024 float4 / 256 threads
        float4 v = xp[threadIdx.x + i * 256];
        s += v.x + v.y + v.z + v.w;
    }
    __shared__ float red[256];
    red[threadIdx.x] = s;
    __syncthreads();
    for (int ofs = 128; ofs > 0; ofs >>= 1) {
        if (threadIdx.x < (unsigned)ofs) red[threadIdx.x] += red[threadIdx.x + ofs];
        __syncthreads();
    }
    if (threadIdx.x == 0) p[blockIdx.x] = red[0] * (1.0f / (float)HWN);
}

// ---------------------------------------------------------------------------
// Kernel 2: per-(b,hw) channel mean & max. Each thread owns 2 consecutive hw
// positions (float2), loops all 256 channels (stride 16KB, coalesced per c).
// Second read of x mostly hits the 192MB L2.
// grid = B * (4096/512) = 256 blocks, 256 threads.
// ---------------------------------------------------------------------------
__global__ __launch_bounds__(256) void k_spatial_stats(const float* __restrict__ x,
                                                       float* __restrict__ avg /*[B*HW]*/,
                                                       float* __restrict__ mx  /*[B*HW]*/) {
    const int b    = blockIdx.x >> 3;
    const int base = ((blockIdx.x & 7) * 512) + threadIdx.x * 2;
    const float* xb = x + (size_t)b * CC * HWN + base;
    float sx = 0.0f, sy = 0.0f;
    float mxv = -INFINITY, myv = -INFINITY;
    for (int c = 0; c < CC; ++c) {
        float2 v = *(const float2*)(xb + (size_t)c * HWN);
        sx += v.x; sy += v.y;
        mxv = fmaxf(mxv, v.x); myv = fmaxf(myv, v.y);
    }
    const int o = b * HWN + base;
    *(float2*)(avg + o) = make_float2(sx * (1.0f / CC), sy * (1.0f / CC));
    *(float2*)(mx  + o) = make_float2(mxv, myv);
}

// ---------------------------------------------------------------------------
// Kernel 3: channel attention, single block of 256 threads (8 waves).
//  Phase A (waves 0..1): h = p @ w1^T + b1 via V_WMMA_F32_16X16X4_F32,
//     M=32 (wave w owns rows 16w..16w+15), N=16, K=256 -> 64 wmma per wave.
//  Phase B: GroupNorm(1 group over CR) + SiLU, one thread per batch row.
//  Phase C: ca = sigmoid(h @ w2^T + b2), thread c computes all 32 batches.
// A-layout (ISA 7.12.2, 32-bit A 16x4): VGPR0={K0|K2}, VGPR1={K1|K3} across
// lane halves; B assumed mirrored (N striped across lanes).
// ---------------------------------------------------------------------------
__global__ __launch_bounds__(256) void k_channel_attn(const float* __restrict__ p,   // [B*C]
                                                      const float* __restrict__ w1,  // [CR*C]
                                                      const float* __restrict__ b1,  // [CR]
                                                      const float* __restrict__ gn_g,
                                                      const float* __restrict__ gn_b,
                                                      const float* __restrict__ w2,  // [C*CR]
                                                      const float* __restrict__ b2,  // [C]
                                                      float* __restrict__ ca) {      // [B*C]
    __shared__ float sP[BB * CC];    // 32 KB
    __shared__ float sW1[CR * CC];   // 16 KB
    __shared__ float sH[BB][CR];     //  2 KB

    const int tid = threadIdx.x;
    for (int i = tid; i < BB * CC; i += 256) sP[i] = p[i];
    for (int i = tid; i < CR * CC; i += 256) sW1[i] = w1[i];
    __syncthreads();

    const int wave = tid >> 5;       // wave-uniform
    const int lane = tid & 31;
    if (wave < 2) {                  // EXEC all-1s inside (uniform branch)
        const int mbase = wave * 16;
        const int n_or_m = lane & 15;          // M row (for A) / N col (for B)
        const int khalf  = (lane >> 4) * 2;    // lanes 0-15 -> K+{0,1}; 16-31 -> K+{2,3}
        const float* arow = &sP[(mbase + n_or_m) * CC + khalf];
        const float* brow = &sW1[n_or_m * CC + khalf];
        v8f acc = {};
        for (int k = 0; k < CC; k += 4) {
            v2f a, b;
            a.x = arow[k];     a.y = arow[k + 1];
            b.x = brow[k];     b.y = brow[k + 1];
            acc = __builtin_amdgcn_wmma_f32_16x16x4_f32(
                false, a, false, b, (short)0, acc, false, false);
        }
        // D layout: VGPR v -> (M = v + 8*(lane>=16), N = lane&15)
        const int n    = lane & 15;
        const int mofs = (lane >> 4) * 8;
        const float bias = b1[n];
        #pragma unroll
        for (int v = 0; v < 8; ++v) sH[mbase + mofs + v][n] = acc[v] + bias;
    }
    __syncthreads();

    if (tid < BB) {                  // GroupNorm(1, CR) + SiLU per batch row
        float mu = 0.0f;
        #pragma unroll
        for (int r = 0; r < CR; ++r) mu += sH[tid][r];
        mu *= (1.0f / CR);
        float var = 0.0f;
        #pragma unroll
        for (int r = 0; r < CR; ++r) { float d = sH[tid][r] - mu; var += d * d; }
        var *= (1.0f / CR);
        const float inv = rsqrtf(var + GN_EPS);
        #pragma unroll
        for (int r = 0; r < CR; ++r) {
            float hv = (sH[tid][r] - mu) * inv * gn_g[r] + gn_b[r];
            sH[tid][r] = hv * sigmoidf_(hv);   // SiLU
        }
    }
    __syncthreads();

    {   // excite + sigmoid: thread c handles column c for all batches
        const int c = tid;
        float wr[CR];
        #pragma unroll
        for (int r = 0; r < CR; ++r) wr[r] = w2[c * CR + r];
        const float bias = b2[c];
        for (int b = 0; b < BB; ++b) {
            float s = bias;
            #pragma unroll
            for (int r = 0; r < CR; ++r) s += sH[b][r] * wr[r];
            ca[b * CC + c] = sigmoidf_(s);
        }
    }
}

// ---------------------------------------------------------------------------
// Kernel 4: 7x7 spatial conv over [avg,max] channels + bias + sigmoid.
// grid = B*H (2048), block = W (64). Trivial FLOPs; neighbors hit cache.
// ---------------------------------------------------------------------------
__global__ __launch_bounds__(64) void k_sa_conv(const float* __restrict__ avg,
                                                const float* __restrict__ mx,
                                                const float* __restrict__ sa_w, // [2*49]
                                                const float* __restrict__ sa_b, // [1]
                                                float* __restrict__ sa) {       // [B*HW]
    const int b = blockIdx.x >> 6;
    const int h = blockIdx.x & 63;
    const int w = threadIdx.x;
    const float* A = avg + (size_t)b * HWN;
    const float* M = mx  + (size_t)b * HWN;
    float s = sa_b[0];
    #pragma unroll
    for (int dy = -3; dy <= 3; ++dy) {
        const int hh = h + dy;
        if (hh < 0 || hh >= HH) continue;
        #pragma unroll
        for (int dx = -3; dx <= 3; ++dx) {
            const int ww = w + dx;
            if (ww < 0 || ww >= WW) continue;
            const int ki = (dy + 3) * 7 + (dx + 3);
            const int o  = hh * WW + ww;
            s += sa_w[ki] * A[o] + sa_w[49 + ki] * M[o];
        }
    }
    sa[b * HWN + h * WW + w] = sigmoidf_(s);
}

// ---------------------------------------------------------------------------
// Kernel 5: out = x * (w0*ca[b,c] + w1*sa[b,hw]); float4 streaming pass.
// grid = B*C*HW/4/256 = 32768, block = 256. softmax(balance) inlined.
// ---------------------------------------------------------------------------
__global__ __launch_bounds__(256) void k_combine(const float* __restrict__ x,
                                                 const float* __restrict__ ca,      // [B*C]
                                                 const float* __restrict__ sa,      // [B*HW]
                                                 const float* __restrict__ balance, // [2]
                                                 float* __restrict__ out) {
    const float bl0 = balance[0], bl1 = balance[1];
    const float mb = fmaxf(bl0, bl1);
    const float e0 = expf(bl0 - mb), e1 = expf(bl1 - mb);
    const float w0 = e0 / (e0 + e1), w1 = e1 / (e0 + e1);

    const size_t i4    = (size_t)blockIdx.x * 256 + threadIdx.x; // float4 index
    const size_t plane = i4 >> 10;        // (b*C + c); 1024 float4 per plane
    const int    pos4  = (int)(i4 & 1023);
    const int    b     = (int)(plane >> 8);

    const float  cav = ca[plane];
    const float4 xv  = ((const float4*)x)[i4];
    const float4 sv  = ((const float4*)(sa + (size_t)b * HWN))[pos4];

    float4 o;
    o.x = xv.x * (w0 * cav + w1 * sv.x);
    o.y = xv.y * (w0 * cav + w1 * sv.y);
    o.z = xv.z * (w0 * cav + w1 * sv.z);
    o.w = xv.w * (w0 * cav + w1 * sv.w);
    ((float4*)out)[i4] = o;
}

// ---------------------------------------------------------------------------
extern "C" void kernel_launch(void* const* d_in, const int* in_sizes, int n_in,
                              void* d_out, int out_size, void* d_ws, size_t ws_size,
                              hipStream_t stream) {
    (void)in_sizes; (void)n_in; (void)out_size; (void)ws_size;
    const float* x       = (const float*)d_in[0];
    const float* w1      = (const float*)d_in[1];
    const float* b1      = (const float*)d_in[2];
    const float* gn_g    = (const float*)d_in[3];
    const float* gn_b    = (const float*)d_in[4];
    const float* w2      = (const float*)d_in[5];
    const float* b2      = (const float*)d_in[6];
    const float* sa_w    = (const float*)d_in[7];
    const float* sa_b    = (const float*)d_in[8];
    const float* balance = (const float*)d_in[9];
    float* out = (float*)d_out;

    // Workspace layout (floats): p[8192] | ca[8192] | avg[131072] | mx[131072] | sa[131072]
    float* ws  = (float*)d_ws;
    float* p   = ws;
    float* ca  = ws + BB * CC;
    float* avg = ca + BB * CC;
    float* mx  = avg + BB * HWN;
    float* sa  = mx + BB * HWN;

    k_plane_mean   <<<BB * CC,            256, 0, stream>>>(x, p);
    k_spatial_stats<<<BB * (HWN / 512),   256, 0, stream>>>(x, avg, mx);
    k_channel_attn <<<1,                  256, 0, stream>>>(p, w1, b1, gn_g, gn_b, w2, b2, ca);
    k_sa_conv      <<<BB * HH,             64, 0, stream>>>(avg, mx, sa_w, sa_b, sa);
    k_combine      <<<(BB * CC * HWN) / (4 * 256), 256, 0, stream>>>(x, ca, sa, balance, out);
}